// FCOS_14104672600443
// MI455X (gfx1250) — compile-verified
//
#include <hip/hip_runtime.h>
#include <stdint.h>

// ---------------- problem constants (match reference) ----------------
#define BATCH     16
#define NCLS      80
#define N_P3      16384
#define N_P4      4096
#define N_P5      1024
#define NTOT      21504          // N_P3+N_P4+N_P5
#define KTOP      1024
#define NSORT     32768          // next pow2 >= NTOT
#define IMG_SZ    1024.0f
#define IOU_THRESH 0.5f
#define SCORE_THRESH 0.3f

// output layout (floats), concatenated flat in reference return order
#define OUT_BX    0
#define OUT_SC    (BATCH*KTOP*4)               // 65536
#define OUT_CI    (OUT_SC + BATCH*KTOP)        // 81920
#define OUT_KEEP  (OUT_CI + BATCH*KTOP)        // 98304; total 114688

typedef __attribute__((ext_vector_type(2))) float v2f;
typedef __attribute__((ext_vector_type(8))) float v8f;

#if __has_builtin(__builtin_amdgcn_wmma_f32_16x16x4_f32)
#define HAVE_WMMA_F32_4 1
#endif

__device__ __forceinline__ unsigned ballot32(bool p) {
#if __has_builtin(__builtin_amdgcn_ballot_w32)
  return __builtin_amdgcn_ballot_w32(p);
#else
  return (unsigned)__ballot(p);
#endif
}

__device__ __forceinline__ float sigm(float x) { return 1.0f / (1.0f + __expf(-x)); }

// ================= Kernel 1: per-level decode ========================
// one thread per (batch, location): argmax over 80 classes, joint score,
// LTRB decode + clip. HBM-bound: ~110MB of cls logits total (~5us floor
// at 23.3 TB/s; fits in 192MB L2). N is a power of two -> templated so
// t/N is a shift, stride is a literal constant.
template <int N, int LOFF, int STRIDE_I>
__global__ __launch_bounds__(256)
void fcos_decode(const float* __restrict__ cls, const float* __restrict__ box,
                 const float* __restrict__ ctr, const float* __restrict__ loc,
                 float* __restrict__ boxesAll, float* __restrict__ scoresAll,
                 int* __restrict__ clsAll)
{
  int t = blockIdx.x * blockDim.x + threadIdx.x;
  if (t >= BATCH * N) return;
  int b = t / N, n = t - b * N;              // N pow2 -> shift/mask
  const float stride = (float)STRIDE_I;

  const float4* cp4 = (const float4*)(cls + (size_t)t * NCLS);
  __builtin_prefetch(box + (size_t)t * 4, 0, 3);   // global_prefetch_b8
  __builtin_prefetch(ctr + t, 0, 3);

  float maxv = -3.4e38f; int maxi = 0;             // first-max (jnp.argmax ties)
  #pragma unroll
  for (int i = 0; i < NCLS / 4; ++i) {
    float4 v = cp4[i];
    if (v.x > maxv) { maxv = v.x; maxi = 4*i + 0; }
    if (v.y > maxv) { maxv = v.y; maxi = 4*i + 1; }
    if (v.z > maxv) { maxv = v.z; maxi = 4*i + 2; }
    if (v.w > maxv) { maxv = v.w; maxi = 4*i + 3; }
  }
  float score = sqrtf(sigm(maxv) * sigm(ctr[t]));

  float lx = loc[2*n + 0], ly = loc[2*n + 1];
  const float4 d4 = *(const float4*)(box + (size_t)t * 4);
  float d0 = fmaxf(d4.x, 0.f), d1 = fmaxf(d4.y, 0.f);
  float d2 = fmaxf(d4.z, 0.f), d3 = fmaxf(d4.w, 0.f);
  float x1 = fminf(fmaxf(lx - stride * d0, 0.f), IMG_SZ);
  float y1 = fminf(fmaxf(ly - stride * d1, 0.f), IMG_SZ);
  float x2 = fminf(fmaxf(lx + stride * d2, 0.f), IMG_SZ);
  float y2 = fminf(fmaxf(ly + stride * d3, 0.f), IMG_SZ);

  size_t g = (size_t)b * NTOT + LOFF + n;
  float* bo = boxesAll + g * 4;
  bo[0] = x1; bo[1] = y1; bo[2] = x2; bo[3] = y2;
  scoresAll[g] = score;
  clsAll[g]    = maxi;
}

// ================= Kernel 2: per-image top-1024 ======================
// One workgroup per image; bitonic sort of 32768 u64 keys entirely in
// 256KB LDS (CDNA5's 320KB/WGP makes this single-pass). Key packs
// (~ordered(score) << 32) | idx  ->  ascending sort == score-desc,
// idx-asc on ties (matches jax.lax.top_k).
__global__ __launch_bounds__(1024)
void fcos_topk(const float* __restrict__ scoresAll, const float* __restrict__ boxesAll,
               const int* __restrict__ clsAll, float* __restrict__ out)
{
  extern __shared__ char dynsmem[];
  unsigned long long* keys = (unsigned long long*)dynsmem;  // 32768 * 8B
  int img = blockIdx.x, tid = threadIdx.x;
  const float* sc = scoresAll + (size_t)img * NTOT;

  for (int i = tid; i < NSORT; i += 1024) {
    unsigned long long k;
    if (i < NTOT) {
      unsigned u = __float_as_uint(sc[i]);
      u = (u & 0x80000000u) ? ~u : (u | 0x80000000u);  // totally-ordered ascending
      k = ((unsigned long long)(~u) << 32) | (unsigned)i;
    } else {
      k = 0xFFFFFFFFFFFFFFFFull;                       // -inf score padding
    }
    keys[i] = k;
  }
  __syncthreads();

  for (unsigned kk = 2; kk <= NSORT; kk <<= 1) {
    for (unsigned j = kk >> 1; j > 0; j >>= 1) {
      for (unsigned i = tid; i < NSORT; i += 1024) {
        unsigned ixj = i ^ j;
        if (ixj > i) {
          unsigned long long a = keys[i], b = keys[ixj];
          bool up = ((i & kk) == 0);
          if ((a > b) == up) { keys[i] = b; keys[ixj] = a; }
        }
      }
      __syncthreads();
    }
  }

  // gather top-1024 (thread t = rank t)
  unsigned idx = (unsigned)(keys[tid] & 0xFFFFFFFFu);
  size_t g = (size_t)img * NTOT + idx;
  size_t o = (size_t)img * KTOP + tid;
  const float* bi = boxesAll + g * 4;
  float* bo = out + OUT_BX + o * 4;
  bo[0] = bi[0]; bo[1] = bi[1]; bo[2] = bi[2]; bo[3] = bi[3];
  out[OUT_SC + o] = scoresAll[g];
  out[OUT_CI + o] = (float)clsAll[g];
}

// ================= Kernel 3: per-image class-spec NMS ================
// LDS: mask[1024][64] u32 (16 bits each, one 16-col tile) 256KB
//      + offset boxes 16KB + area 4KB + score 4KB + red 1KB + supp 128B.
// Union term per 16x16 tile via V_WMMA_F32_16X16X4_F32 rank-2 sum,
// intersection on VALU (co-executes), ballot_w32 builds the bitmask.
// Threshold test is division-free: iou>=t  <=>  uni>0 && inter>=t*uni.
#define MASK_BYTES (KTOP * 64 * 4)
#define NMS_LDS_BYTES (MASK_BYTES + KTOP*4*4 + KTOP*4 + KTOP*4 + 256*4 + 32*4)

__global__ __launch_bounds__(256)
void fcos_nms(float* __restrict__ out)
{
  extern __shared__ char dynsmem[];
  unsigned* mask = (unsigned*)dynsmem;
  float*    bxs  = (float*)(dynsmem + MASK_BYTES);
  float*    area = (float*)(dynsmem + MASK_BYTES + KTOP*4*4);
  float*    scs  = (float*)(dynsmem + MASK_BYTES + KTOP*4*4 + KTOP*4);
  float*    red  = (float*)(dynsmem + MASK_BYTES + KTOP*4*4 + KTOP*4 + KTOP*4);
  volatile unsigned* supp =
      (volatile unsigned*)(dynsmem + MASK_BYTES + KTOP*4*4 + KTOP*4 + KTOP*4 + 256*4);

  int img = blockIdx.x, tid = threadIdx.x;
  const float* inBx = out + OUT_BX + (size_t)img * KTOP * 4;

  // ---- async global->LDS stage of the 16KB of raw boxes (no VGPR hop),
  //      4 floats per lane per issue; drain with s_wait_asynccnt 0.
  for (int i = tid * 4; i < KTOP * 4; i += 256 * 4) {
    unsigned ldsA = (unsigned)(unsigned long long)(void*)(bxs + i);  // flat[31:0] = LDS offset
    unsigned long long ga = (unsigned long long)(const void*)(inBx + i);
    asm volatile("global_load_async_to_lds_b128 %0, %1, off"
                 :: "v"(ldsA), "v"(ga) : "memory");
  }
  asm volatile("s_wait_asynccnt 0x0" ::: "memory");
  __syncthreads();

  // ---- reduce max_coord over all 4096 coords ----
  float local = 0.0f;                      // boxes clipped to [0,1024]
  for (int i = tid; i < KTOP * 4; i += 256) local = fmaxf(local, bxs[i]);
  red[tid] = local;
  __syncthreads();
  for (int s = 128; s > 0; s >>= 1) {
    if (tid < s) red[tid] = fmaxf(red[tid], red[tid + s]);
    __syncthreads();
  }
  float maxc = red[0];
  __syncthreads();

  // ---- class offset, area, score ----
  for (int row = tid; row < KTOP; row += 256) {
    float off = out[OUT_CI + (size_t)img * KTOP + row] * (maxc + 1.0f);
    float x1 = bxs[row*4+0] + off, y1 = bxs[row*4+1] + off;
    float x2 = bxs[row*4+2] + off, y2 = bxs[row*4+3] + off;
    bxs[row*4+0] = x1; bxs[row*4+1] = y1; bxs[row*4+2] = x2; bxs[row*4+3] = y2;
    area[row] = (x2 - x1) * (y2 - y1);
    scs[row]  = out[OUT_SC + (size_t)img * KTOP + row];
  }
  __syncthreads();

  // ---- IoU bitmask: 64x64 tiles of 16x16; one wave per tile ----
  int wave = tid >> 5, lane = tid & 31;
  int nWaves = blockDim.x >> 5;
  int half = lane >> 4, l15 = lane & 31 & 15;
  for (int tile = wave; tile < 64 * 64; tile += nWaves) {
    int ti = tile >> 6, tj = tile & 63;
    int n = tj * 16 + l15;                 // D layout: N = lane&15 for both halves
    int mbase = ti * 16 + (half ? 8 : 0);  // D layout: M = r (+8 for lanes 16-31)

    // A(16x4) = [area_i | 1 | 0 | 0], B(4x16) = [1 ; area_j ; 0 ; 0]
    float aI = area[ti * 16 + l15];
    float aJ = area[n];
    float one = (half == 0) ? 1.0f : 0.0f; // lanes 16-31 carry K=2,3 (zeros)
    v2f Av; Av.x = (half == 0) ? aI : 0.0f; Av.y = one;
    v2f Bv; Bv.x = one; Bv.y = (half == 0) ? aJ : 0.0f;

    v8f D;
#ifdef HAVE_WMMA_F32_4
    v8f Cz = {0.f, 0.f, 0.f, 0.f, 0.f, 0.f, 0.f, 0.f};
    // D[m][n] = area_m + area_n on the matrix pipe
    D = __builtin_amdgcn_wmma_f32_16x16x4_f32(
        false, Av, false, Bv, (short)0, Cz, false, false);
#else
    #pragma unroll
    for (int r = 0; r < 8; ++r) D[r] = area[mbase + r] + aJ;
#endif

    float nx1 = bxs[n*4+0], ny1 = bxs[n*4+1], nx2 = bxs[n*4+2], ny2 = bxs[n*4+3];
    #pragma unroll
    for (int r = 0; r < 8; ++r) {
      int m = mbase + r;
      float mx1 = bxs[m*4+0], my1 = bxs[m*4+1], mx2 = bxs[m*4+2], my2 = bxs[m*4+3];
      float iw = fmaxf(fminf(mx2, nx2) - fmaxf(mx1, nx1), 0.0f);
      float ih = fmaxf(fminf(my2, ny2) - fmaxf(my1, ny1), 0.0f);
      float inter = iw * ih;
      float uni = D[r] - inter;             // area_m + area_n - inter
      // iou >= thr  <=>  uni > 0 && inter >= thr*uni  (NaN/0-union -> false)
      bool hit = (uni > 0.0f) && (inter >= IOU_THRESH * uni);
      unsigned bal = ballot32(hit);
      if (lane == 0) {
        mask[(ti*16 + r)     * 64 + tj] = bal & 0xFFFFu;  // rows ti*16+r
        mask[(ti*16 + 8 + r) * 64 + tj] = bal >> 16;      // rows ti*16+8+r
      }
    }
  }
  __syncthreads();

  // ---- greedy scan: single wave, lane L owns suppression bits 32L..32L+31
  if (tid < 32) {
    supp[lane] = 0u;
    float* keepOut = out + OUT_KEEP + (size_t)img * KTOP;
    for (int i = 0; i < KTOP; ++i) {
      unsigned w = supp[i >> 5];                       // broadcast LDS read
      bool keep = (((w >> (i & 31)) & 1u) == 0u) && (scs[i] > SCORE_THRESH);
      if (keep) {
        unsigned bits = mask[i*64 + 2*lane] | (mask[i*64 + 2*lane + 1] << 16);
        supp[lane] |= bits;                            // in-order LDS per wave
      }
      if (lane == 0) keepOut[i] = keep ? 1.0f : 0.0f;
    }
  }
}

// ========================= host launcher =============================
extern "C" void kernel_launch(void* const* d_in, const int* in_sizes, int n_in,
                              void* d_out, int out_size, void* d_ws, size_t ws_size,
                              hipStream_t stream) {
  const float* cls3 = (const float*)d_in[0];
  const float* box3 = (const float*)d_in[1];
  const float* ctr3 = (const float*)d_in[2];
  const float* loc3 = (const float*)d_in[3];
  const float* cls4 = (const float*)d_in[4];
  const float* box4 = (const float*)d_in[5];
  const float* ctr4 = (const float*)d_in[6];
  const float* loc4 = (const float*)d_in[7];
  const float* cls5 = (const float*)d_in[8];
  const float* box5 = (const float*)d_in[9];
  const float* ctr5 = (const float*)d_in[10];
  const float* loc5 = (const float*)d_in[11];
  float* out = (float*)d_out;

  // workspace: boxesAll [B,NTOT,4] | scoresAll [B,NTOT] | clsAll [B,NTOT]  (~8.3MB)
  float* boxesAll  = (float*)d_ws;
  float* scoresAll = boxesAll + (size_t)BATCH * NTOT * 4;
  int*   clsAll    = (int*)(scoresAll + (size_t)BATCH * NTOT);

  fcos_decode<N_P3, 0, 8><<<(BATCH * N_P3 + 255) / 256, 256, 0, stream>>>(
      cls3, box3, ctr3, loc3, boxesAll, scoresAll, clsAll);
  fcos_decode<N_P4, N_P3, 16><<<(BATCH * N_P4 + 255) / 256, 256, 0, stream>>>(
      cls4, box4, ctr4, loc4, boxesAll, scoresAll, clsAll);
  fcos_decode<N_P5, N_P3 + N_P4, 32><<<(BATCH * N_P5 + 255) / 256, 256, 0, stream>>>(
      cls5, box5, ctr5, loc5, boxesAll, scoresAll, clsAll);

  fcos_topk<<<BATCH, 1024, (size_t)NSORT * 8, stream>>>(scoresAll, boxesAll, clsAll, out);

  fcos_nms<<<BATCH, 256, (size_t)NMS_LDS_BYTES, stream>>>(out);
}